// GMIC_67851893342775
// MI455X (gfx1250) — compile-verified
//
#include <hip/hip_runtime.h>
#include <math.h>

// ---------------- constants ----------------
#define BB      2
#define HH      2944
#define WW      1920
#define CAMH    92
#define CAMW    60
#define CROP    256
#define KSEL    6
#define NCLS    2
#define TOPT    110                 // int(0.02*92*60)
#define PHW     85                  // CAMH-8+1
#define PWW     53                  // CAMW-8+1
#define NPIX    (CAMH*CAMW)         // 5520

typedef float v2f __attribute__((ext_vector_type(2)));
typedef float v8f __attribute__((ext_vector_type(8)));

static inline int cdiv(long long a, long long b) { return (int)((a + b - 1) / b); }

// ---------------- WMMA conv 3x3 stride 2, SAME(pad_lo=0,pad_hi=1), ReLU ----------------
// One wave per block. Tile: 16 output channels (M) x 16 output columns (N).
// K permutation: tap-major (9 taps), channel-minor (chunks of 4 input channels).
// A fragment (16x4 f32): lane m = tid&15; regs hold K = {2*hi, 2*hi+1} of the chunk.
// B fragment: col n = tid&15; same K mapping. C/D: reg r -> row r (lanes 0-15) / r+8 (16-31).
// Inner ci loop is software-pipelined: next chunk's 4 loads issue before current WMMA.
__global__ __launch_bounds__(32) void wmma_conv3x3_s2_relu(
    const float* __restrict__ X, const float* __restrict__ Wt, float* __restrict__ Y,
    int Cin, int Hin, int Win, int Cout, int Hout, int Wout)
{
    const int xtiles = (Wout + 15) >> 4;
    const int co0 = (blockIdx.x / xtiles) * 16;
    const int ox0 = (blockIdx.x % xtiles) * 16;
    const int oy  = blockIdx.y;
    const int n   = blockIdx.z;
    const int tid = threadIdx.x;
    const int l16 = tid & 15;
    const int hi  = tid >> 4;

    const size_t HWin   = (size_t)Hin * Win;
    const size_t HWin4  = 4 * HWin;
    const int    Ktot   = Cin * 9;
    const int    nchunk = Cin >> 2;
    const float* Xn     = X + (size_t)n * Cin * HWin;
    const float* wbase  = Wt + (size_t)(co0 + l16) * Ktot + (size_t)(2 * hi) * 9;

    v8f c = {0.f,0.f,0.f,0.f,0.f,0.f,0.f,0.f};

    const int ox = ox0 + l16;

    #pragma unroll
    for (int ky = 0; ky < 3; ++ky) {
        const int iy = oy * 2 + ky;            // wave-uniform
        if (iy >= Hin) continue;               // scalar branch; skipped taps contribute 0
        #pragma unroll
        for (int kx = 0; kx < 3; ++kx) {
            const int ix  = ox * 2 + kx;       // per-lane
            const bool ok = (ix < Win);
            const int  ixc = ok ? ix : (Win - 1);
            const float fv = ok ? 1.f : 0.f;   // zero out-of-range columns (no EXEC divergence)

            const float* wp = wbase + (ky * 3 + kx);
            const float* xp = Xn + (size_t)iy * Win + ixc + (size_t)(2 * hi) * HWin;
            const float* xq = xp + HWin;

            // prologue: chunk 0 loads
            float a0 = wp[0];                  // W[co, ci+2*hi,   ky, kx]
            float a1 = wp[9];                  // W[co, ci+2*hi+1, ky, kx]
            float b0 = xp[0];                  // X[ci+2*hi,   iy, ix]
            float b1 = xq[0];                  // X[ci+2*hi+1, iy, ix]

            for (int i = 0; i < nchunk - 1; ++i) {
                wp += 36;                      // 4 channels * 9 taps
                xp += HWin4;
                xq += HWin4;
                // issue next chunk's loads before consuming current registers
                float na0 = wp[0];
                float na1 = wp[9];
                float nb0 = xp[0];
                float nb1 = xq[0];
                v2f a;  a.x  = a0;       a.y  = a1;
                v2f bf; bf.x = b0 * fv;  bf.y = b1 * fv;
                c = __builtin_amdgcn_wmma_f32_16x16x4_f32(false, a, false, bf,
                                                          (short)0, c, false, false);
                a0 = na0; a1 = na1; b0 = nb0; b1 = nb1;
            }
            // epilogue: last chunk
            {
                v2f a;  a.x  = a0;       a.y  = a1;
                v2f bf; bf.x = b0 * fv;  bf.y = b1 * fv;
                c = __builtin_amdgcn_wmma_f32_16x16x4_f32(false, a, false, bf,
                                                          (short)0, c, false, false);
            }
        }
    }

    if (ox < Wout) {
        #pragma unroll
        for (int r = 0; r < 8; ++r) {
            int m = r + 8 * hi;
            float v = c[r];
            v = v > 0.f ? v : 0.f;
            Y[(((size_t)n * Cout + co0 + m) * Hout + oy) * Wout + ox] = v;
        }
    }
}

// ---------------- direct conv for Cin==1 (K=9), stride 2, ReLU ----------------
__global__ void conv3x3_s2_relu_c1(
    const float* __restrict__ X, const float* __restrict__ Wt, float* __restrict__ Y,
    int N, int Cout, int Hin, int Win, int Hout, int Wout)
{
    size_t idx = (size_t)blockIdx.x * blockDim.x + threadIdx.x;
    size_t total = (size_t)N * Cout * Hout * Wout;
    if (idx >= total) return;
    int ox = (int)(idx % Wout); size_t t = idx / Wout;
    int oy = (int)(t % Hout);   t /= Hout;
    int co = (int)(t % Cout);   int n = (int)(t / Cout);
    const float* w  = Wt + (size_t)co * 9;
    const float* Xn = X + (size_t)n * Hin * Win;
    float s = 0.f;
    #pragma unroll
    for (int ky = 0; ky < 3; ++ky) {
        int iy = oy * 2 + ky;
        if (iy >= Hin) continue;
        #pragma unroll
        for (int kx = 0; kx < 3; ++kx) {
            int ix = ox * 2 + kx;
            if (ix >= Win) continue;
            s = __builtin_fmaf(Xn[(size_t)iy * Win + ix], w[ky * 3 + kx], s);
        }
    }
    Y[idx] = s > 0.f ? s : 0.f;
}

// ---------------- saliency: 1x1 conv (256->2) + sigmoid; also combined = sum over cls ------
__global__ void saliency_kernel(const float* __restrict__ Hg, const float* __restrict__ SW,
                                float* __restrict__ sal_out, float* __restrict__ combined)
{
    int idx = blockIdx.x * blockDim.x + threadIdx.x;     // B*NPIX
    if (idx >= BB * NPIX) return;
    int pix = idx % NPIX;
    int b   = idx / NPIX;
    const float* base = Hg + ((size_t)b * 256) * NPIX + pix;
    float s0 = 0.f, s1 = 0.f;
    for (int cch = 0; cch < 256; ++cch) {
        float v = base[(size_t)cch * NPIX];
        s0 = __builtin_fmaf(v, SW[cch],       s0);
        s1 = __builtin_fmaf(v, SW[256 + cch], s1);
    }
    float g0 = 1.f / (1.f + expf(-s0));
    float g1 = 1.f / (1.f + expf(-s1));
    sal_out[(size_t)(b * NCLS + 0) * NPIX + pix] = g0;
    sal_out[(size_t)(b * NCLS + 1) * NPIX + pix] = g1;
    combined[(size_t)b * NPIX + pix] = g0 + g1;
}

// ---------------- mean of top-110 per (b,cls) ----------------
__global__ __launch_bounds__(256) void topk_mean_kernel(const float* __restrict__ sal,
                                                        float* __restrict__ yglobal)
{
    __shared__ float buf[NPIX];
    __shared__ float rmax[256];
    __shared__ int   ridx[256];
    __shared__ float ssum;
    const float* src = sal + (size_t)blockIdx.x * NPIX;
    for (int i = threadIdx.x; i < NPIX; i += 256) buf[i] = src[i];
    if (threadIdx.x == 0) ssum = 0.f;
    __syncthreads();
    for (int it = 0; it < TOPT; ++it) {
        float best = -INFINITY; int bi = 0x7fffffff;
        for (int i = threadIdx.x; i < NPIX; i += 256) {
            float v = buf[i];
            if (v > best) { best = v; bi = i; }
        }
        rmax[threadIdx.x] = best; ridx[threadIdx.x] = bi;
        __syncthreads();
        for (int s = 128; s > 0; s >>= 1) {
            if (threadIdx.x < s) {
                bool take = rmax[threadIdx.x + s] > rmax[threadIdx.x] ||
                            (rmax[threadIdx.x + s] == rmax[threadIdx.x] &&
                             ridx[threadIdx.x + s] < ridx[threadIdx.x]);
                if (take) { rmax[threadIdx.x] = rmax[threadIdx.x + s];
                            ridx[threadIdx.x] = ridx[threadIdx.x + s]; }
            }
            __syncthreads();
        }
        if (threadIdx.x == 0) { ssum += rmax[0]; buf[ridx[0]] = -INFINITY; }
        __syncthreads();
    }
    if (threadIdx.x == 0) yglobal[blockIdx.x] = ssum / (float)TOPT;
}

// ---------------- bilinear resize (align corners) 92x60 -> 2944x1920 ----------------
__global__ void resize_kernel(const float* __restrict__ sal, float* __restrict__ out)
{
    size_t idx = (size_t)blockIdx.x * blockDim.x + threadIdx.x;
    size_t total = (size_t)BB * NCLS * HH * WW;
    if (idx >= total) return;
    int ox = (int)(idx % WW); size_t t = idx / WW;
    int oy = (int)(t % HH);
    int m  = (int)(t / HH);                  // b*NCLS+cls
    const float sy = (float)(CAMH - 1) / (float)(HH - 1);
    const float sx = (float)(CAMW - 1) / (float)(WW - 1);
    float fy = oy * sy, fx = ox * sx;
    int y0 = (int)floorf(fy); int y1 = y0 + 1; if (y1 > CAMH - 1) y1 = CAMH - 1;
    int x0 = (int)floorf(fx); int x1 = x0 + 1; if (x1 > CAMW - 1) x1 = CAMW - 1;
    float wy = fy - (float)y0, wx = fx - (float)x0;
    const float* mp = sal + (size_t)m * NPIX;
    float a = mp[y0 * CAMW + x0], b = mp[y0 * CAMW + x1];
    float c = mp[y1 * CAMW + x0], d = mp[y1 * CAMW + x1];
    float top = a * (1.f - wx) + b * wx;
    float bot = c * (1.f - wx) + d * wx;
    out[idx] = top * (1.f - wy) + bot * wy;
}

// ---------------- greedy K-crop selection (one block per batch) ----------------
__global__ __launch_bounds__(256) void crop_select_kernel(const float* __restrict__ combined,
                                                          int* __restrict__ pos)
{
    __shared__ float comb[NPIX];
    __shared__ float mask[NPIX];
    __shared__ float bval[256];
    __shared__ int   bidx[256];
    int b = blockIdx.x;
    for (int i = threadIdx.x; i < NPIX; i += 256) {
        comb[i] = combined[(size_t)b * NPIX + i];
        mask[i] = 1.f;
    }
    __syncthreads();
    for (int k = 0; k < KSEL; ++k) {
        float best = -INFINITY; int bi = 0x7fffffff;
        for (int w = threadIdx.x; w < PHW * PWW; w += 256) {
            int py = w / PWW, px = w % PWW;
            float s = 0.f;
            for (int dy = 0; dy < 8; ++dy)
                for (int dx = 0; dx < 8; ++dx) {
                    int j = (py + dy) * CAMW + (px + dx);
                    s = __builtin_fmaf(comb[j], mask[j], s);
                }
            if (s > best || (s == best && w < bi)) { best = s; bi = w; }
        }
        bval[threadIdx.x] = best; bidx[threadIdx.x] = bi;
        __syncthreads();
        for (int s2 = 128; s2 > 0; s2 >>= 1) {
            if (threadIdx.x < s2) {
                bool take = bval[threadIdx.x + s2] > bval[threadIdx.x] ||
                            (bval[threadIdx.x + s2] == bval[threadIdx.x] &&
                             bidx[threadIdx.x + s2] < bidx[threadIdx.x]);
                if (take) { bval[threadIdx.x] = bval[threadIdx.x + s2];
                            bidx[threadIdx.x] = bidx[threadIdx.x + s2]; }
            }
            __syncthreads();
        }
        int wi = bidx[0]; int py = wi / PWW; int px = wi % PWW;
        if (threadIdx.x < 64) {
            int dy = threadIdx.x >> 3, dx = threadIdx.x & 7;
            mask[(py + dy) * CAMW + (px + dx)] = 0.f;
        }
        if (threadIdx.x == 0) {
            pos[(b * KSEL + k) * 2 + 0] = py * 32;   // scale H/CAMH = 32 exactly
            pos[(b * KSEL + k) * 2 + 1] = px * 32;   // scale W/CAMW = 32 exactly
        }
        __syncthreads();
    }
}

// ---------------- crop extraction ----------------
__global__ void crop_extract(const float* __restrict__ X, const int* __restrict__ pos,
                             float* __restrict__ crops)
{
    size_t idx = (size_t)blockIdx.x * blockDim.x + threadIdx.x;
    size_t total = (size_t)BB * KSEL * CROP * CROP;
    if (idx >= total) return;
    int x = (int)(idx % CROP); size_t t = idx / CROP;
    int y = (int)(t % CROP);
    int bk = (int)(t / CROP);
    int b  = bk / KSEL;
    int py = pos[bk * 2 + 0], px = pos[bk * 2 + 1];
    crops[idx] = X[((size_t)b * HH + py + y) * WW + (px + x)];
}

// ---------------- spatial max over 16x16 ----------------
__global__ void spatial_max_kernel(const float* __restrict__ A, float* __restrict__ out)
{
    int idx = blockIdx.x * blockDim.x + threadIdx.x;     // 12*512
    if (idx >= BB * KSEL * 512) return;
    const float* p = A + (size_t)idx * 256;
    float m = p[0];
    for (int i = 1; i < 256; ++i) m = fmaxf(m, p[i]);
    out[idx] = m;
}

// ---------------- attention MIL + classifier (single block, deterministic) ----------------
__global__ __launch_bounds__(256) void attention_cls_kernel(
    const float* __restrict__ hc,  const float* __restrict__ V,
    const float* __restrict__ U,   const float* __restrict__ aw,
    const float* __restrict__ cw,  const float* __restrict__ cb,
    float* __restrict__ ylocal)
{
    __shared__ float garr[BB * KSEL * 128];
    __shared__ float logits[BB * KSEL];
    __shared__ float attn[BB * KSEL];
    __shared__ float z[BB * 512];
    int tid = threadIdx.x;

    for (int task = tid; task < BB * KSEL * 128; task += 256) {
        int d = task & 127; int bk = task >> 7;
        const float* h = hc + (size_t)bk * 512;
        float sv = 0.f, su = 0.f;
        for (int c = 0; c < 512; ++c) {
            float hv = h[c];
            sv = __builtin_fmaf(hv, V[c * 128 + d], sv);
            su = __builtin_fmaf(hv, U[c * 128 + d], su);
        }
        garr[task] = tanhf(sv) * (1.f / (1.f + expf(-su))) * aw[d];
    }
    __syncthreads();
    if (tid < BB * KSEL) {
        float s = 0.f;
        for (int d = 0; d < 128; ++d) s += garr[tid * 128 + d];
        logits[tid] = s;
    }
    __syncthreads();
    if (tid < BB) {
        float mx = -INFINITY;
        for (int k = 0; k < KSEL; ++k) mx = fmaxf(mx, logits[tid * KSEL + k]);
        float s = 0.f;
        for (int k = 0; k < KSEL; ++k) { float e = expf(logits[tid * KSEL + k] - mx);
                                         attn[tid * KSEL + k] = e; s += e; }
        for (int k = 0; k < KSEL; ++k) attn[tid * KSEL + k] /= s;
    }
    __syncthreads();
    for (int task = tid; task < BB * 512; task += 256) {
        int c = task & 511; int b = task >> 9;
        float s = 0.f;
        for (int k = 0; k < KSEL; ++k)
            s = __builtin_fmaf(attn[b * KSEL + k], hc[(size_t)(b * KSEL + k) * 512 + c], s);
        z[task] = s;
    }
    __syncthreads();
    if (tid < BB * NCLS) {
        int cls = tid % NCLS; int b = tid / NCLS;
        float s = cb[cls];
        for (int c = 0; c < 512; ++c)
            s = __builtin_fmaf(z[b * 512 + c], cw[c * NCLS + cls], s);
        ylocal[tid] = 1.f / (1.f + expf(-s));
    }
}

// ---------------- host orchestration ----------------
extern "C" void kernel_launch(void* const* d_in, const int* in_sizes, int n_in,
                              void* d_out, int out_size, void* d_ws, size_t ws_size,
                              hipStream_t stream) {
    (void)in_sizes; (void)n_in; (void)out_size; (void)ws_size;
    const float* x    = (const float*)d_in[0];
    const float* gk1  = (const float*)d_in[1];
    const float* gk2  = (const float*)d_in[2];
    const float* gk3  = (const float*)d_in[3];
    const float* gk4  = (const float*)d_in[4];
    const float* gk5  = (const float*)d_in[5];
    const float* salw = (const float*)d_in[6];
    const float* lk1  = (const float*)d_in[7];
    const float* lk2  = (const float*)d_in[8];
    const float* lk3  = (const float*)d_in[9];
    const float* lk4  = (const float*)d_in[10];
    const float* attV = (const float*)d_in[11];
    const float* attU = (const float*)d_in[12];
    const float* attw = (const float*)d_in[13];
    const float* clsw = (const float*)d_in[14];
    const float* clsb = (const float*)d_in[15];

    float* out = (float*)d_out;
    float* ws  = (float*)d_ws;

    // d_out layout: y_global[4] | y_local[4] | saliency[22080] | resized[22609920]
    float* o_yglob = out;
    float* o_yloc  = out + 4;
    float* o_sal   = out + 8;
    float* o_rsz   = out + 8 + BB * NCLS * NPIX;

    // workspace layout (floats)
    float* bufA     = ws;                       // 45,219,840
    float* bufB     = ws + 45219840;            // 22,609,920
    float* act5     = ws + 67829760;            //  2,826,240
    float* combined = ws + 70656000;            //     11,040
    int*   pos      = (int*)(ws + 70667040);    //         24 ints (64 slots reserved)
    float* crops    = ws + 70667104;            //    786,432
    float* hcrops   = ws + 71453536;            //      6,144

    // ---- global backbone ----
    {   // conv1: 1->16, 2944x1920 -> 1472x960
        long long tot = (long long)BB * 16 * 1472 * 960;
        conv3x3_s2_relu_c1<<<cdiv(tot, 256), 256, 0, stream>>>(
            x, gk1, bufA, BB, 16, HH, WW, 1472, 960);
    }
    {   // conv2: 16->32, 1472x960 -> 736x480
        dim3 g((32 / 16) * cdiv(480, 16), 736, BB);
        wmma_conv3x3_s2_relu<<<g, 32, 0, stream>>>(bufA, gk2, bufB, 16, 1472, 960, 32, 736, 480);
    }
    {   // conv3: 32->64, 736x480 -> 368x240
        dim3 g((64 / 16) * cdiv(240, 16), 368, BB);
        wmma_conv3x3_s2_relu<<<g, 32, 0, stream>>>(bufB, gk3, bufA, 32, 736, 480, 64, 368, 240);
    }
    {   // conv4: 64->128, 368x240 -> 184x120
        dim3 g((128 / 16) * cdiv(120, 16), 184, BB);
        wmma_conv3x3_s2_relu<<<g, 32, 0, stream>>>(bufA, gk4, bufB, 64, 368, 240, 128, 184, 120);
    }
    {   // conv5: 128->256, 184x120 -> 92x60
        dim3 g((256 / 16) * cdiv(60, 16), 92, BB);
        wmma_conv3x3_s2_relu<<<g, 32, 0, stream>>>(bufB, gk5, act5, 128, 184, 120, 256, 92, 60);
    }

    // ---- saliency, y_global, resized map ----
    saliency_kernel<<<cdiv(BB * NPIX, 256), 256, 0, stream>>>(act5, salw, o_sal, combined);
    topk_mean_kernel<<<BB * NCLS, 256, 0, stream>>>(o_sal, o_yglob);
    resize_kernel<<<cdiv((long long)BB * NCLS * HH * WW, 256), 256, 0, stream>>>(o_sal, o_rsz);

    // ---- crop selection + extraction ----
    crop_select_kernel<<<BB, 256, 0, stream>>>(combined, pos);
    crop_extract<<<cdiv((long long)BB * KSEL * CROP * CROP, 256), 256, 0, stream>>>(x, pos, crops);

    // ---- local backbone on 12 crops ----
    {   // lconv1: 1->32, 256 -> 128
        long long tot = (long long)BB * KSEL * 32 * 128 * 128;
        conv3x3_s2_relu_c1<<<cdiv(tot, 256), 256, 0, stream>>>(
            crops, lk1, bufA, BB * KSEL, 32, CROP, CROP, 128, 128);
    }
    {   // lconv2: 32->64, 128 -> 64
        dim3 g((64 / 16) * cdiv(64, 16), 64, BB * KSEL);
        wmma_conv3x3_s2_relu<<<g, 32, 0, stream>>>(bufA, lk2, bufB, 32, 128, 128, 64, 64, 64);
    }
    {   // lconv3: 64->128, 64 -> 32
        dim3 g((128 / 16) * cdiv(32, 16), 32, BB * KSEL);
        wmma_conv3x3_s2_relu<<<g, 32, 0, stream>>>(bufB, lk3, bufA, 64, 64, 64, 128, 32, 32);
    }
    {   // lconv4: 128->512, 32 -> 16
        dim3 g((512 / 16) * cdiv(16, 16), 16, BB * KSEL);
        wmma_conv3x3_s2_relu<<<g, 32, 0, stream>>>(bufA, lk4, bufB, 128, 32, 32, 512, 16, 16);
    }

    // ---- pooling + attention + classifier ----
    spatial_max_kernel<<<cdiv(BB * KSEL * 512, 256), 256, 0, stream>>>(bufB, hcrops);
    attention_cls_kernel<<<1, 256, 0, stream>>>(hcrops, attV, attU, attw, clsw, clsb, o_yloc);
}